// SSKernelNPLR_87273735454841
// MI455X (gfx1250) — compile-verified
//
#include <hip/hip_runtime.h>

// SSKernelNPLR fused kernel for gfx1250 (MI455X).
// One block per h: Cauchy einsum via V_WMMA_F32_16X16X4_F32 (fp32 matrix core),
// Woodbury correction, then 1024-pt inverse complex FFT in LDS for the irfft.
// Hot-loop reciprocals use v_rcp_f32 (TRANS pipe) so they co-execute with WMMA.

#define SSK_H   256
#define SSK_N   64
#define SSK_L   2048
#define SSK_LH  1025
#define SSK_NF  1024

typedef __attribute__((ext_vector_type(2))) float v2f;
typedef __attribute__((ext_vector_type(8))) float v8f;

#if __has_builtin(__builtin_amdgcn_wmma_f32_16x16x4_f32)
#define SSK_USE_WMMA 1
#else
#define SSK_USE_WMMA 0
#endif

__device__ __forceinline__ float ssk_pi() { return 3.14159265358979323846f; }

// Fast reciprocal: single v_rcp_f32 (~1 ulp), fine for fp32-grade results.
__device__ __forceinline__ float ssk_rcp(float x) {
#if __has_builtin(__builtin_amdgcn_rcpf)
    return __builtin_amdgcn_rcpf(x);
#else
    return 1.0f / x;
#endif
}

__global__ __launch_bounds__(128)
void ssk_fused_kernel(const float* __restrict__ w_re, const float* __restrict__ w_im,
                      const float* __restrict__ B_re, const float* __restrict__ B_im,
                      const float* __restrict__ P_re, const float* __restrict__ P_im,
                      const float* __restrict__ C_re, const float* __restrict__ C_im,
                      const float* __restrict__ log_dt,
                      float* __restrict__ out)
{
    // A matrix (16 x 64): rows 0-3 = Re(v_j), rows 4-7 = Im(v_j), rows 8-15 = 0.
    // v_j = Bc[a]*Cc[b], j = 2a+b, Bc=[B;P], Cc=[C;conj(P)].
    __shared__ float  sA[16 * SSK_N];   // 4 KB
    __shared__ float2 sW[SSK_N];        // w*dt, 0.5 KB
    __shared__ float2 sKF[SSK_LH];      // k_f spectrum, 8.2 KB
    __shared__ float  sFr[SSK_NF];      // FFT re, 4 KB
    __shared__ float  sFi[SSK_NF];      // FFT im, 4 KB

    const int h   = blockIdx.x;
    const int tid = threadIdx.x;          // 0..127
    const float dt = __expf(log_dt[h]);

    if (tid < SSK_N) {
        sW[tid] = make_float2(w_re[h * SSK_N + tid] * dt,
                              w_im[h * SSK_N + tid] * dt);
    }
    for (int idx = tid; idx < 16 * SSK_N; idx += 128) {
        const int row = idx >> 6;
        const int n   = idx & 63;
        float val = 0.0f;
        if (row < 8) {
            const int j = row & 3, a = j >> 1, b = j & 1;
            const int o = h * SSK_N + n;
            const float xr = a ? P_re[o] : B_re[o];
            const float xi = a ? P_im[o] : B_im[o];
            const float yr = b ? P_re[o] : C_re[o];
            const float yi = b ? -P_im[o] : C_im[o];   // Q = conj(P)
            const float vr = xr * yr - xi * yi;
            const float vi = xr * yi + xi * yr;
            val = (row < 4) ? vr : vi;
        }
        sA[row * SSK_N + n] = val;
    }
    __syncthreads();

    // ---- Cauchy einsum over 16-wide l tiles, 65 tiles, 4 waves round-robin ----
    const int wave  = tid >> 5;   // wave32
    const int lane  = tid & 31;
    const int m     = lane & 15;  // A row (lanes) / D column (l within tile)
    const int halfk = lane >> 4;  // K-half select for A/B fragments

    for (int t = wave; t < 65; t += 4) {
        const int l = t * 16 + m;
        // omega = e^{-2pi i l / L}; z = 2(1-omega)/(1+omega) = i * zim (pure imag)
        float s, c;
        __sincosf(-ssk_pi() * (float)l * (1.0f / 1024.0f), &s, &c);
        const float den = fmaxf(1.0f + c, 1e-30f);     // keep l=1024 pole finite
        const float zim = -2.0f * s * ssk_rcp(den);

        float r0r = 0, r0i = 0, r1r = 0, r1i = 0;
        float r2r = 0, r2i = 0, r3r = 0, r3i = 0;
        const bool active = (lane < 16) && (l < SSK_LH);

#if SSK_USE_WMMA
        v8f d1 = {0.f,0.f,0.f,0.f,0.f,0.f,0.f,0.f};   // A x Re(cauchy)
        v8f d2 = {0.f,0.f,0.f,0.f,0.f,0.f,0.f,0.f};   // A x Im(cauchy)
#pragma unroll
        for (int kk = 0; kk < 16; ++kk) {
            const int n = kk * 4 + halfk * 2;
            // A fragment: lane (m, halfk) holds A[m][n], A[m][n+1]
            v2f a;
            a.x = sA[m * SSK_N + n];
            a.y = sA[m * SSK_N + n + 1];
            // B fragment: lane holds cauchy[n][l], cauchy[n+1][l]
            const float2 w0 = sW[n];
            const float2 w1 = sW[n + 1];
            const float dr0 = -w0.x, di0 = zim - w0.y;
            const float inv0 = ssk_rcp(dr0 * dr0 + di0 * di0);  // v_rcp_f32
            const float dr1 = -w1.x, di1 = zim - w1.y;
            const float inv1 = ssk_rcp(dr1 * dr1 + di1 * di1);  // v_rcp_f32
            v2f bre, bim;
            bre.x = dr0 * inv0;  bre.y = dr1 * inv1;
            bim.x = -di0 * inv0; bim.y = -di1 * inv1;
            d1 = __builtin_amdgcn_wmma_f32_16x16x4_f32(false, a, false, bre,
                                                       (short)0, d1, false, false);
            d2 = __builtin_amdgcn_wmma_f32_16x16x4_f32(false, a, false, bim,
                                                       (short)0, d2, false, false);
        }
        if (active) {
            // r_re[j] = D1[j] - D2[j+4]; r_im[j] = D2[j] + D1[j+4]; then * dt
            r0r = (d1[0] - d2[4]) * dt;  r0i = (d2[0] + d1[4]) * dt;
            r1r = (d1[1] - d2[5]) * dt;  r1i = (d2[1] + d1[5]) * dt;
            r2r = (d1[2] - d2[6]) * dt;  r2i = (d2[2] + d1[6]) * dt;
            r3r = (d1[3] - d2[7]) * dt;  r3i = (d2[3] + d1[7]) * dt;
        }
#else
        if (active) {
            for (int n = 0; n < SSK_N; ++n) {
                const float2 w = sW[n];
                const float dr = -w.x, di = zim - w.y;
                const float inv = ssk_rcp(dr * dr + di * di);
                const float cr = dr * inv, ci = -di * inv;
                const float v0r = sA[0*SSK_N+n], v0i = sA[4*SSK_N+n];
                const float v1r = sA[1*SSK_N+n], v1i = sA[5*SSK_N+n];
                const float v2r = sA[2*SSK_N+n], v2i = sA[6*SSK_N+n];
                const float v3r = sA[3*SSK_N+n], v3i = sA[7*SSK_N+n];
                r0r += v0r*cr - v0i*ci;  r0i += v0r*ci + v0i*cr;
                r1r += v1r*cr - v1i*ci;  r1i += v1r*ci + v1i*cr;
                r2r += v2r*cr - v2i*ci;  r2i += v2r*ci + v2i*cr;
                r3r += v3r*cr - v3i*ci;  r3i += v3r*ci + v3i*cr;
            }
            r0r *= dt; r0i *= dt; r1r *= dt; r1i *= dt;
            r2r *= dt; r2i *= dt; r3r *= dt; r3i *= dt;
        }
#endif

        if (active) {
            // Woodbury: kf = r00 - r01*r10/(1+r11), then * 2/(1+omega) = 1 + i*zim/2
            const float dnr = 1.0f + r3r, dni = r3i;
            const float dinv = ssk_rcp(dnr * dnr + dni * dni);
            const float nr = r1r * r2r - r1i * r2i;
            const float ni = r1r * r2i + r1i * r2r;
            const float qr = (nr * dnr + ni * dni) * dinv;
            const float qi = (ni * dnr - nr * dni) * dinv;
            const float kr = r0r - qr, ki = r0i - qi;
            const float gi = 0.5f * zim;
            sKF[l] = make_float2(kr - ki * gi, ki + kr * gi);
        }
    }
    __syncthreads();

    // ---- irfft(2048) via 1024-pt inverse complex FFT ----
    // pack: E[k]=(X[k]+conj(X[1024-k]))/2, O[k]=(X[k]-conj(X[1024-k]))/2 * e^{+i pi k/1024}
    // Z[k]=E+iO, stored bit-reversed.
    for (int k = tid; k < SSK_NF; k += 128) {
        const float2 Xk = sKF[k];
        const float2 Xm = sKF[SSK_NF - k];
        const float er = 0.5f * (Xk.x + Xm.x);
        const float ei = 0.5f * (Xk.y - Xm.y);
        const float tr = 0.5f * (Xk.x - Xm.x);
        const float ti = 0.5f * (Xk.y + Xm.y);
        float s, c;
        __sincosf(ssk_pi() * (float)k * (1.0f / 1024.0f), &s, &c);
        const float orr = tr * c - ti * s;
        const float oi  = tr * s + ti * c;
        const int idx = (int)(__brev((unsigned)k) >> 22);   // bitrev10
        sFr[idx] = er - oi;
        sFi[idx] = ei + orr;
    }
    __syncthreads();

    // radix-2 DIT, inverse twiddles (+ sign), 10 stages
    for (int st = 0; st < 10; ++st) {
        const int half = 1 << st;
        const float astep = ssk_pi() * ssk_rcp((float)half);  // pi / half
        for (int b = tid; b < 512; b += 128) {
            const int j  = b & (half - 1);
            const int i0 = ((b >> st) << (st + 1)) + j;
            const int i1 = i0 + half;
            float sn, cs;
            __sincosf(astep * (float)j, &sn, &cs);
            const float ur = sFr[i0], ui = sFi[i0];
            const float vr = sFr[i1], vi = sFi[i1];
            const float wr = vr * cs - vi * sn;
            const float wi = vr * sn + vi * cs;
            sFr[i0] = ur + wr;  sFi[i0] = ui + wi;
            sFr[i1] = ur - wr;  sFi[i1] = ui - wi;
        }
        __syncthreads();
    }

    // y[m] = x[2m] + i x[2m+1]; normalize by 1/1024
    const float scale = 1.0f / 1024.0f;
    float* __restrict__ o = out + (size_t)h * SSK_L;
    for (int mm = tid; mm < SSK_NF; mm += 128) {
        o[2 * mm]     = sFr[mm] * scale;
        o[2 * mm + 1] = sFi[mm] * scale;
    }
}

extern "C" void kernel_launch(void* const* d_in, const int* in_sizes, int n_in,
                              void* d_out, int out_size, void* d_ws, size_t ws_size,
                              hipStream_t stream) {
    (void)in_sizes; (void)n_in; (void)out_size; (void)d_ws; (void)ws_size;
    const float* w_re   = (const float*)d_in[0];
    const float* w_im   = (const float*)d_in[1];
    const float* B_re   = (const float*)d_in[2];
    const float* B_im   = (const float*)d_in[3];
    const float* P_re   = (const float*)d_in[4];
    const float* P_im   = (const float*)d_in[5];
    const float* C_re   = (const float*)d_in[6];
    const float* C_im   = (const float*)d_in[7];
    const float* log_dt = (const float*)d_in[8];
    // d_in[9] = L (2048), compile-time constant here
    ssk_fused_kernel<<<dim3(SSK_H), dim3(128), 0, stream>>>(
        w_re, w_im, B_re, B_im, P_re, P_im, C_re, C_im, log_dt, (float*)d_out);
}